// EuclideanCodebook_69252052680738
// MI455X (gfx1250) — compile-verified
//
#include <hip/hip_runtime.h>
#include <hip/hip_bf16.h>

typedef __attribute__((ext_vector_type(2))) float v2f;
typedef __attribute__((ext_vector_type(8))) float v8f;
typedef __attribute__((ext_vector_type(4))) float v4f;

#define D_DIM 64
#define TILE_N 32          // codes per LDS stage (2 WMMA tiles)
#define ROWS_PER_WAVE 16
#define WAVES_PER_BLOCK 8
#define ROWS_PER_BLOCK (ROWS_PER_WAVE * WAVES_PER_BLOCK)  // 128

// ---------------------------------------------------------------------------
// Kernel 1: zero cluster_size / embed_sum regions of d_out, compute e2[K]
// ---------------------------------------------------------------------------
__global__ void vq_init_kernel(const float* __restrict__ embed,
                               float* __restrict__ cluster_size,  // [K]
                               float* __restrict__ embed_sum,     // [K*D]
                               float* __restrict__ e2,            // [K] scratch
                               int K) {
    int tid = blockIdx.x * blockDim.x + threadIdx.x;
    if (tid < K * D_DIM) embed_sum[tid] = 0.0f;
    if (tid < K) {
        cluster_size[tid] = 0.0f;
        const float* row = embed + (size_t)tid * D_DIM;
        float acc = 0.0f;
#pragma unroll 8
        for (int j = 0; j < D_DIM; ++j) acc += row[j] * row[j];
        e2[tid] = acc;
    }
}

// ---------------------------------------------------------------------------
// Kernel 2: fused distance-GEMM (WMMA f32 16x16x4, dual accumulator chains)
//           + streaming argmin + gather + scatter
// ---------------------------------------------------------------------------
__global__ __launch_bounds__(256) void vq_codebook_kernel(
    const float* __restrict__ x,       // [N, 64]
    const float* __restrict__ embed,   // [K, 64]
    const float* __restrict__ e2,      // [K]
    float* __restrict__ quantize,      // [N, 64]
    float* __restrict__ out_ind,       // [N]  (index stored as float)
    float* __restrict__ cluster_size,  // [K]
    float* __restrict__ embed_sum,     // [K, 64]
    int K) {

    __shared__ float sB[TILE_N * D_DIM];   // 32x64 f32 code tile = 8 KB

    const int lane = threadIdx.x & 31;
    const int wave = threadIdx.x >> 5;
    const int rowBase = blockIdx.x * ROWS_PER_BLOCK + wave * ROWS_PER_WAVE;

    // ---- Preload A fragments for this wave's 16 rows of x ----
    // A (16x4 f32): lanes 0-15 hold M=lane, K={k0,k0+1}; lanes 16-31 hold
    // M=lane-16, K={k0+2,k0+3}.  One v2f per K-step of 4; D=64 -> 16 steps.
    const int arow = lane & 15;
    const int kp   = (lane >> 4) << 1;      // 0 or 2
    v2f afrag[16];
#pragma unroll
    for (int s = 0; s < 16; ++s) {
        afrag[s] = *(const v2f*)(x + (size_t)(rowBase + arow) * D_DIM + s * 4 + kp);
    }

    float bestS[8];
    int   bestI[8];
#pragma unroll
    for (int r = 0; r < 8; ++r) { bestS[r] = 3.4e38f; bestI[r] = 0; }

    const int ncol = lane & 15;             // this lane's code column in a tile

    for (int n0 = 0; n0 < K; n0 += TILE_N) {
        // Stage embed[n0 : n0+32, :] into LDS: 2048 floats, 2 float4/thread.
        __syncthreads();
        {
            const v4f* src = (const v4f*)(embed + (size_t)n0 * D_DIM);
            ((v4f*)sB)[threadIdx.x]       = src[threadIdx.x];
            ((v4f*)sB)[threadIdx.x + 256] = src[threadIdx.x + 256];
        }
        __syncthreads();

        if (n0 + TILE_N < K)
            __builtin_prefetch(embed + (size_t)(n0 + TILE_N) * D_DIM + lane * 64, 0, 0);

        // B (4x16 f32): lanes 0-15 hold N=lane, K={k0,k0+1}; lanes 16-31 hold
        // N=lane-16, K={k0+2,k0+3}.  Two independent accumulator chains so the
        // matrix pipe always has a ready WMMA (no serial D->C stall).
        v8f acc0 = {};
        v8f acc1 = {};
#pragma unroll
        for (int s = 0; s < 16; ++s) {
            v2f b0 = *(const v2f*)(sB + ncol * D_DIM + s * 4 + kp);
            v2f b1 = *(const v2f*)(sB + (ncol + 16) * D_DIM + s * 4 + kp);
            acc0 = __builtin_amdgcn_wmma_f32_16x16x4_f32(
                false, afrag[s], false, b0, (short)0, acc0, false, false);
            acc1 = __builtin_amdgcn_wmma_f32_16x16x4_f32(
                false, afrag[s], false, b1, (short)0, acc1, false, false);
        }

        // score = e2[n] - 2*(x . e)   (||x||^2 dropped: constant per row)
        const float e2a = e2[n0 + ncol];
        const float e2b = e2[n0 + 16 + ncol];
        const int ca = n0 + ncol;
        const int cb = n0 + 16 + ncol;
#pragma unroll
        for (int r = 0; r < 8; ++r) {
            float s0 = __builtin_fmaf(-2.0f, acc0[r], e2a);
            if (s0 < bestS[r]) { bestS[r] = s0; bestI[r] = ca; }
            float s1 = __builtin_fmaf(-2.0f, acc1[r], e2b);
            if (s1 < bestS[r]) { bestS[r] = s1; bestI[r] = cb; }
        }
    }

    // ---- argmin reduction across the 16 lanes of each half-wave ----
    // VGPR slot r: lanes 0-15 carry row r, lanes 16-31 carry row r+8.
#pragma unroll
    for (int r = 0; r < 8; ++r) {
        float s = bestS[r];
        int   i = bestI[r];
#pragma unroll
        for (int off = 8; off >= 1; off >>= 1) {
            float so = __shfl_xor(s, off, 32);
            int   io = __shfl_xor(i, off, 32);
            if (so < s || (so == s && io < i)) { s = so; i = io; }
        }
        bestS[r] = s;
        bestI[r] = i;
    }

    // ---- per-row outputs: index + cluster_size (one lane per half) ----
    if (lane == 0 || lane == 16) {
        const int mBase = (lane >> 4) << 3;
#pragma unroll
        for (int r = 0; r < 8; ++r) {
            const int m = mBase + r;
            out_ind[rowBase + m] = (float)bestI[r];
            atomicAdd(&cluster_size[bestI[r]], 1.0f);
        }
    }

    // ---- gather quantize rows; scatter embed_sum ----
    const int col = lane * 2;
#pragma unroll 1
    for (int m = 0; m < 16; ++m) {
        const int r = m & 7;
        const int srcLane = (m < 8) ? 0 : 16;
        const int idx = __shfl(bestI[r], srcLane, 32);

        const v2f ev = *(const v2f*)(embed + (size_t)idx * D_DIM + col);
        *(v2f*)(quantize + (size_t)(rowBase + m) * D_DIM + col) = ev;

        const v2f xv = *(const v2f*)(x + (size_t)(rowBase + m) * D_DIM + col);
        atomicAdd(&embed_sum[(size_t)idx * D_DIM + col],     xv.x);
        atomicAdd(&embed_sum[(size_t)idx * D_DIM + col + 1], xv.y);
    }
}

// ---------------------------------------------------------------------------
extern "C" void kernel_launch(void* const* d_in, const int* in_sizes, int n_in,
                              void* d_out, int out_size, void* d_ws, size_t ws_size,
                              hipStream_t stream) {
    const float* x     = (const float*)d_in[0];
    const float* embed = (const float*)d_in[1];

    const int N = in_sizes[0] / D_DIM;   // 131072
    const int K = in_sizes[1] / D_DIM;   // 4096

    // d_out layout: quantize[N*64] | embed_ind[N] | cluster_size[K] | embed_sum[K*64]
    float* quantize     = (float*)d_out;
    float* out_ind      = quantize + (size_t)N * D_DIM;
    float* cluster_size = out_ind + N;
    float* embed_sum    = cluster_size + K;

    float* e2 = (float*)d_ws;            // K floats of scratch

    {
        const int total = K * D_DIM;
        const int blocks = (total + 255) / 256;
        vq_init_kernel<<<blocks, 256, 0, stream>>>(embed, cluster_size, embed_sum, e2, K);
    }
    {
        const int blocks = N / ROWS_PER_BLOCK;   // 1024
        vq_codebook_kernel<<<blocks, 256, 0, stream>>>(
            x, embed, e2, quantize, out_ind, cluster_size, embed_sum, K);
    }
}